// SpeakerEmbeddingSystem_53446573031779
// MI455X (gfx1250) — compile-verified
//
#include <hip/hip_runtime.h>
#include <hip/hip_bf16.h>
#include <math.h>

typedef float v2f __attribute__((ext_vector_type(2)));
typedef float v8f __attribute__((ext_vector_type(8)));

#define D 256
#define BLK_ROWS 128          // rows of the batch per block (8 waves x 16 rows)
#define KCHUNK 32             // K-slice staged in LDS per iteration
#define NKCHUNK 8             // 256 / 32
#define WSTRIDE 264           // 256 + 8 dword pad -> halves of the wave hit disjoint LDS banks
#define XSTRIDE 36            // 32 + 4 dword pad, keeps rows 16B aligned

// ---------------------------------------------------------------------------
// Kernel 1: build the 4x256 "base" table.
// base[s] = LN(relu(combined_s @ w1 + b1)) @ w2 + b2 + 0.3*emo_b
// combined_s depends only on speaker s (4 values) and constants.
// ---------------------------------------------------------------------------
__global__ __launch_bounds__(256)
void speaker_base_table_kernel(const float* __restrict__ emb_table,
                               const float* __restrict__ pitch_w,  const float* __restrict__ pitch_b,
                               const float* __restrict__ energy_w, const float* __restrict__ energy_b,
                               const float* __restrict__ rate_w,   const float* __restrict__ rate_b,
                               const float* __restrict__ accent_table,
                               const int*   __restrict__ accent_level,
                               const float* __restrict__ w1, const float* __restrict__ b1,
                               const float* __restrict__ ln_g, const float* __restrict__ ln_b,
                               const float* __restrict__ w2, const float* __restrict__ b2,
                               const float* __restrict__ emo_b,
                               float* __restrict__ base_out)
{
    __shared__ float cmb[512];
    __shared__ float h[256];
    __shared__ float hn[256];

    const int s = blockIdx.x;      // speaker 0..3
    const int t = threadIdx.x;     // 0..255 (one output channel per thread)

    const float PITCH[4]  = {0.60f, 0.50f, 1.10f, 0.95f};   // /200 pre-applied
    const float ENERGY[4] = {0.80f, 0.60f, 0.85f, 0.70f};
    const float RATE[4]   = {1.10f, 0.90f, 1.15f, 1.00f};
    const int lvl = accent_level[0];

    for (int k = t; k < 512; k += 256) {
        float v;
        if      (k < 256) v = emb_table[s * 256 + k];
        else if (k < 320) v = fmaf(PITCH[s],  pitch_w[k - 256],  pitch_b[k - 256]);
        else if (k < 384) v = fmaf(ENERGY[s], energy_w[k - 320], energy_b[k - 320]);
        else if (k < 448) v = fmaf(RATE[s],   rate_w[k - 384],   rate_b[k - 384]);
        else              v = accent_table[lvl * 64 + (k - 448)];
        cmb[k] = v;
    }
    __syncthreads();

    float a = b1[t];
    for (int k = 0; k < 512; ++k) a = fmaf(cmb[k], w1[k * 256 + t], a);
    h[t] = fmaxf(a, 0.0f);
    __syncthreads();

    // LayerNorm (redundant per-thread reduction over 256 LDS values; trivial cost)
    float mu = 0.0f;
    for (int k = 0; k < 256; ++k) mu += h[k];
    mu *= (1.0f / 256.0f);
    float var = 0.0f;
    for (int k = 0; k < 256; ++k) { float d = h[k] - mu; var = fmaf(d, d, var); }
    var *= (1.0f / 256.0f);
    hn[t] = (h[t] - mu) * rsqrtf(var + 1e-5f) * ln_g[t] + ln_b[t];
    __syncthreads();

    float o = fmaf(0.3f, emo_b[t], b2[t]);
    for (int k = 0; k < 256; ++k) o = fmaf(hn[k], w2[k * 256 + t], o);
    base_out[s * 256 + t] = o;
}

// ---------------------------------------------------------------------------
// Kernel 2: out = normalize(base[sid] + 0.3 * (emotion @ emo_w))
// WMMA f32 16x16x4. One wave per 16-row x 256-col tile (16 accumulator tiles).
// ---------------------------------------------------------------------------
__global__ __launch_bounds__(256)
void speaker_fused_gemm_kernel(const float* __restrict__ emo,
                               const float* __restrict__ emo_w,
                               const int*   __restrict__ speaker_id,
                               const float* __restrict__ base,
                               float* __restrict__ out)
{
    __shared__ float sW[KCHUNK * WSTRIDE];     //  33,792 B
    __shared__ float sX[BLK_ROWS * XSTRIDE];   //  18,432 B
    __shared__ float sBase[4 * 256];           //   4,096 B   (total 56,320 B)

    const int tid  = threadIdx.x;
    const int lane = tid & 31;
    const int wave = tid >> 5;          // 0..7
    const int lrow = lane & 15;         // N index inside a 16-wide tile
    const int hi   = (lane >= 16) ? 1 : 0;
    const int wrb  = wave * 16;         // wave's row base inside the block
    const int blockRow = blockIdx.x * BLK_ROWS;

    for (int i = tid; i < 4 * 256; i += 256) sBase[i] = base[i];

    v8f acc[16];
#pragma unroll
    for (int ct = 0; ct < 16; ++ct)
#pragma unroll
        for (int j = 0; j < 8; ++j) acc[ct][j] = 0.0f;

    for (int kc = 0; kc < NKCHUNK; ++kc) {
        const int kbase = kc * KCHUNK;
        __syncthreads();
        // Stage X tile: 128 rows x 32 k, coalesced 128b loads (8 float4 per row)
#pragma unroll
        for (int i = 0; i < 4; ++i) {
            int idx = tid + 256 * i;
            int row = idx >> 3, seg = idx & 7;
            float4 v = *reinterpret_cast<const float4*>(
                &emo[(size_t)(blockRow + row) * D + kbase + seg * 4]);
            *reinterpret_cast<float4*>(&sX[row * XSTRIDE + seg * 4]) = v;
        }
        // Stage W slice: 32 k-rows x 256 cols
#pragma unroll
        for (int i = 0; i < 8; ++i) {
            int idx = tid + 256 * i;
            int row = idx >> 6, seg = idx & 63;
            float4 v = *reinterpret_cast<const float4*>(
                &emo_w[(size_t)(kbase + row) * D + seg * 4]);
            *reinterpret_cast<float4*>(&sW[row * WSTRIDE + seg * 4]) = v;
        }
        __syncthreads();

#pragma unroll
        for (int kk = 0; kk < KCHUNK / 4; ++kk) {
            // A fragment (16x4 f32): lanes 0-15 hold K=0,1 ; lanes 16-31 hold K=2,3
            const int k4 = kk * 4 + hi * 2;
            float2 af = *reinterpret_cast<const float2*>(
                &sX[(wrb + lrow) * XSTRIDE + k4]);
            v2f av; av[0] = af.x; av[1] = af.y;
#pragma unroll
            for (int ct = 0; ct < 16; ++ct) {
                // B fragment (4x16 f32): lanes 0-15 rows K=0,1 ; lanes 16-31 rows K=2,3
                v2f bv;
                bv[0] = sW[(k4)     * WSTRIDE + ct * 16 + lrow];
                bv[1] = sW[(k4 + 1) * WSTRIDE + ct * 16 + lrow];
                acc[ct] = __builtin_amdgcn_wmma_f32_16x16x4_f32(
                    false, av, false, bv, (short)0, acc[ct], false, false);
            }
        }
    }

    // Epilogue: + base[sid], scale 0.3, L2-normalize per row, store.
    // D layout: VGPR v -> row (wrb + v) for lanes 0-15, row (wrb + v + 8) for lanes 16-31.
    int   sid[8];
    float sq[8];
#pragma unroll
    for (int v = 0; v < 8; ++v) {
        sid[v] = speaker_id[blockRow + wrb + hi * 8 + v];
        sq[v] = 0.0f;
    }
#pragma unroll
    for (int ct = 0; ct < 16; ++ct)
#pragma unroll
        for (int v = 0; v < 8; ++v) {
            float val = fmaf(0.3f, acc[ct][v], sBase[sid[v] * 256 + ct * 16 + lrow]);
            acc[ct][v] = val;
            sq[v] = fmaf(val, val, sq[v]);
        }
    float inv[8];
#pragma unroll
    for (int v = 0; v < 8; ++v) {
        float s = sq[v];
        s += __shfl_xor(s, 1, 16);   // halves of the wave hold disjoint rows
        s += __shfl_xor(s, 2, 16);
        s += __shfl_xor(s, 4, 16);
        s += __shfl_xor(s, 8, 16);
        inv[v] = 1.0f / fmaxf(sqrtf(s), 1e-12f);
    }
#pragma unroll
    for (int ct = 0; ct < 16; ++ct)
#pragma unroll
        for (int v = 0; v < 8; ++v) {
            size_t row = (size_t)(blockRow + wrb + hi * 8 + v);
            out[row * D + ct * 16 + lrow] = acc[ct][v] * inv[v];
        }
}

extern "C" void kernel_launch(void* const* d_in, const int* in_sizes, int n_in,
                              void* d_out, int out_size, void* d_ws, size_t ws_size,
                              hipStream_t stream) {
    const int*   speaker_id   = (const int*)  d_in[0];
    const float* emotion      = (const float*)d_in[1];
    const int*   accent_level = (const int*)  d_in[2];
    const float* emb_table    = (const float*)d_in[3];
    const float* pitch_w      = (const float*)d_in[4];
    const float* pitch_b      = (const float*)d_in[5];
    const float* energy_w     = (const float*)d_in[6];
    const float* energy_b     = (const float*)d_in[7];
    const float* rate_w       = (const float*)d_in[8];
    const float* rate_b       = (const float*)d_in[9];
    const float* accent_table = (const float*)d_in[10];
    const float* w1           = (const float*)d_in[11];
    const float* b1           = (const float*)d_in[12];
    const float* ln_g         = (const float*)d_in[13];
    const float* ln_b         = (const float*)d_in[14];
    const float* w2           = (const float*)d_in[15];
    const float* b2           = (const float*)d_in[16];
    const float* emo_w        = (const float*)d_in[17];
    const float* emo_b        = (const float*)d_in[18];

    float* base = (float*)d_ws;           // 4 x 256 floats
    float* outp = (float*)d_out;

    const int B = in_sizes[0];            // 131072

    speaker_base_table_kernel<<<4, 256, 0, stream>>>(
        emb_table, pitch_w, pitch_b, energy_w, energy_b, rate_w, rate_b,
        accent_table, accent_level, w1, b1, ln_g, ln_b, w2, b2, emo_b, base);

    speaker_fused_gemm_kernel<<<B / BLK_ROWS, 256, 0, stream>>>(
        emotion, emo_w, speaker_id, base, outp);
}